// GINE_34935263986010
// MI455X (gfx1250) — compile-verified
//
#include <hip/hip_runtime.h>

// GINEConv x2 on gfx1250 (MI455X), wave32.
//   edge phase : wave-per-edge gather + fused rank-1 edge MLP + relu + f32 L2 atomics
//   node phase : fp32 GEMM via V_WMMA_F32_16X16X4_F32, A-tile staged in LDS
typedef __attribute__((ext_vector_type(2))) float v2f;
typedef __attribute__((ext_vector_type(8))) float v8f;

#define DIM 128

// ---------------------------------------------------------------------------
// Edge kernel: one wave32 per edge; lane L handles floats [4L, 4L+4).
// msg = relu(feat[src] + w * We + be); agg[dst] += msg  (atomic f32)
// ---------------------------------------------------------------------------
__global__ __launch_bounds__(256) void gine_edge_kernel(
    const float* __restrict__ feat,
    const long long* __restrict__ srcI,
    const long long* __restrict__ dstI,
    const float* __restrict__ ew,
    const float* __restrict__ We,
    const float* __restrict__ be,
    float* __restrict__ agg,
    int nEdges)
{
    int gtid = blockIdx.x * blockDim.x + threadIdx.x;
    int e    = gtid >> 5;                // wave32: one edge per wave
    if (e >= nEdges) return;
    int lane = threadIdx.x & 31;

    long long s = srcI[e];
    long long d = dstI[e];
    float     w = ew[e];

    // Edge-MLP row chunk (L2/L0 resident, 512B total)
    float4 we4 = ((const float4*)We)[lane];
    float4 be4 = ((const float4*)be)[lane];
    // Gather source-node features: 512B contiguous per edge, float4 per lane
    float4 xv  = ((const float4*)(feat + (size_t)s * DIM))[lane];

    float4 m;
    m.x = fmaxf(xv.x + w * we4.x + be4.x, 0.0f);
    m.y = fmaxf(xv.y + w * we4.y + be4.y, 0.0f);
    m.z = fmaxf(xv.z + w * we4.z + be4.z, 0.0f);
    m.w = fmaxf(xv.w + w * we4.w + be4.w, 0.0f);

    float* a = agg + (size_t)d * DIM + lane * 4;
    atomicAdd(a + 0, m.x);
    atomicAdd(a + 1, m.y);
    atomicAdd(a + 2, m.z);
    atomicAdd(a + 3, m.w);
}

// ---------------------------------------------------------------------------
// GEMM kernel: out = act((X + AGG) @ W + bias)
// Block = 256 threads = 8 waves; one 16-row M-tile per block; wave wv owns
// N-columns [16*wv, 16*wv+16). K=128 stepped by 4 with v_wmma_f32_16x16x4_f32.
//
// A 16x4 f32 fragment layout (ISA 7.12.2): lane L row M=L&15;
//   VGPR0 = K (lanes 0-15) / K+2 (lanes 16-31), VGPR1 = K+1 / K+3.
// C/D 16x16 f32: VGPR r -> M=r (lanes 0-15) / M=r+8 (lanes 16-31), N=lane&15.
// ---------------------------------------------------------------------------
__global__ __launch_bounds__(256) void gine_gemm_kernel(
    const float* __restrict__ X,
    const float* __restrict__ AGG,
    const float* __restrict__ W,      // [K=128][N=128] row-major
    const float* __restrict__ bias,   // [128]
    float* __restrict__ out,
    int doRelu)
{
    __shared__ float Atile[16][DIM + 4];   // +4 pad: conflict-free column reads

    const int tid   = threadIdx.x;
    const int mtile = blockIdx.x;

    // Cooperatively stage A = X + AGG (16 x 128 floats = 8KB) into LDS.
    const float* Xp = X   + (size_t)mtile * 16 * DIM;
    const float* Gp = AGG + (size_t)mtile * 16 * DIM;
    {
        int i   = tid * 4;               // 256 threads * 2 float4 = 2048 floats
        int row = i >> 7;
        int col = i & 127;
        float4 xv = *(const float4*)(Xp + row * DIM + col);
        float4 gv = *(const float4*)(Gp + row * DIM + col);
        Atile[row][col + 0] = xv.x + gv.x;
        Atile[row][col + 1] = xv.y + gv.y;
        Atile[row][col + 2] = xv.z + gv.z;
        Atile[row][col + 3] = xv.w + gv.w;
        i  += 1024;
        row = i >> 7;
        col = i & 127;
        xv = *(const float4*)(Xp + row * DIM + col);
        gv = *(const float4*)(Gp + row * DIM + col);
        Atile[row][col + 0] = xv.x + gv.x;
        Atile[row][col + 1] = xv.y + gv.y;
        Atile[row][col + 2] = xv.z + gv.z;
        Atile[row][col + 3] = xv.w + gv.w;
    }
    __syncthreads();

    const int wv   = tid >> 5;           // wave id 0..7 -> N-tile
    const int lane = tid & 31;
    const int mloc = lane & 15;          // A row within tile
    const int kb   = (lane >> 4) << 1;   // K sub-block 0 or 2
    const int n    = wv * 16 + (lane & 15);

    v8f c = {0.f, 0.f, 0.f, 0.f, 0.f, 0.f, 0.f, 0.f};

    #pragma unroll 4
    for (int k = 0; k < DIM; k += 4) {
        v2f a, b;
        a.x = Atile[mloc][k + kb];
        a.y = Atile[mloc][k + kb + 1];
        const float* wp = W + (size_t)(k + kb) * DIM + n;   // B: K=k+kb / k+kb+1
        b.x = wp[0];
        b.y = wp[DIM];
        c = __builtin_amdgcn_wmma_f32_16x16x4_f32(
                /*neg_a=*/false, a, /*neg_b=*/false, b,
                /*c_mod=*/(short)0, c, /*reuse_a=*/false, /*reuse_b=*/false);
    }

    const float bi    = bias[n];
    const int   mbase = mtile * 16 + ((lane >> 4) << 3);
    #pragma unroll
    for (int r = 0; r < 8; ++r) {
        float v = c[r] + bi;
        if (doRelu) v = fmaxf(v, 0.0f);
        out[(size_t)(mbase + r) * DIM + n] = v;
    }
}

// ---------------------------------------------------------------------------
extern "C" void kernel_launch(void* const* d_in, const int* in_sizes, int n_in,
                              void* d_out, int out_size, void* d_ws, size_t ws_size,
                              hipStream_t stream)
{
    const float*     x   = (const float*)d_in[0];
    const long long* ei  = (const long long*)d_in[1];   // int64 edge_index [2, E]
    const float*     ew  = (const float*)d_in[2];
    const float*     We1 = (const float*)d_in[3];
    const float*     be1 = (const float*)d_in[4];
    const float*     W1  = (const float*)d_in[5];
    const float*     b1  = (const float*)d_in[6];
    const float*     We2 = (const float*)d_in[7];
    const float*     be2 = (const float*)d_in[8];
    const float*     W2  = (const float*)d_in[9];
    const float*     b2  = (const float*)d_in[10];
    float*           out = (float*)d_out;

    const int nNodes = in_sizes[0] / DIM;   // 50000
    const int nEdges = in_sizes[2];         // 800000
    const long long* src = ei;
    const long long* dst = ei + nEdges;

    float* agg = (float*)d_ws;                         // 25.6 MB, reused per layer
    float* h   = agg + (size_t)nNodes * DIM;           // 25.6 MB hidden activations
    const size_t aggBytes = (size_t)nNodes * DIM * sizeof(float);

    const int edgeBlocks = (nEdges * 32 + 255) / 256;  // wave per edge
    const int mTiles     = nNodes / 16;                // 3125

    // ---- Layer 1 ----
    hipMemsetAsync(agg, 0, aggBytes, stream);
    gine_edge_kernel<<<edgeBlocks, 256, 0, stream>>>(x, src, dst, ew, We1, be1, agg, nEdges);
    gine_gemm_kernel<<<mTiles, 256, 0, stream>>>(x, agg, W1, b1, h, /*relu=*/1);

    // ---- Layer 2 ----
    hipMemsetAsync(agg, 0, aggBytes, stream);
    gine_edge_kernel<<<edgeBlocks, 256, 0, stream>>>(h, src, dst, ew, We2, be2, agg, nEdges);
    gine_gemm_kernel<<<mTiles, 256, 0, stream>>>(h, agg, W2, b2, out, /*relu=*/0);
}